// EncodeLayer_25245817766349
// MI455X (gfx1250) — compile-verified
//
#include <hip/hip_runtime.h>

typedef float v2f __attribute__((ext_vector_type(2)));
typedef float v8f __attribute__((ext_vector_type(8)));
typedef unsigned int u32x4 __attribute__((ext_vector_type(4)));
typedef int i32x4 __attribute__((ext_vector_type(4)));
typedef int i32x8 __attribute__((ext_vector_type(8)));

// D = A(16x4 f32) * B(4x16 f32) + C(16x16 f32)  -- native fp32 matrix path on CDNA5
__device__ __forceinline__ v8f wmma4(v2f a, v2f b, v8f c) {
  return __builtin_amdgcn_wmma_f32_16x16x4_f32(false, a, false, b, (short)0, c,
                                               false, false);
}

#define DM   1024   // d_model
#define SEQ  1024
#define BS   8
#define NH   16
#define DK   64

// ---------------------------------------------------------------------------
// TDM: one-shot 64 KB (16384 f32, contiguous) global -> LDS DMA.
// D# per CDNA5 ISA ch.8: group0 = {count, lds_addr, global_addr, type=2},
// group1 = {data_size=4B, tensor_dim0 = tile_dim0 = 16384, dim1=1, stride0}.
// Issued by one wave; completion via TENSORcnt.
// ---------------------------------------------------------------------------
#if __has_builtin(__builtin_amdgcn_tensor_load_to_lds) && \
    __has_builtin(__builtin_amdgcn_s_wait_tensorcnt)
#define USE_TDM 1
#else
#define USE_TDM 0
#endif

__device__ __forceinline__ void tdm_copy_16k_f32(const float* gsrc,
                                                 void* lds_dst) {
#if USE_TDM
  const unsigned long long ga = (unsigned long long)(uintptr_t)gsrc;
  const unsigned int ldsoff = (unsigned int)(uintptr_t)lds_dst;  // LDS offset
  u32x4 g0;
  g0.x = 1u;                                       // count=1 (valid D#)
  g0.y = ldsoff;                                   // lds_addr (bytes)
  g0.z = (unsigned int)ga;                         // global_addr[31:0]
  g0.w = (unsigned int)(ga >> 32) | (2u << 30);    // global_addr[56:32]|type=2
  i32x8 g1;
  g1[0] = 0x00020000;    // data_size=2 (4B); workgroup_mask=0; no pad/iterate
  g1[1] = 0x40000000;    // tensor_dim0[15:0]=0x4000 -> bits[31:16]
  g1[2] = 0x00010000;    // tensor_dim0[31:16]=0 | tensor_dim1=1
  g1[3] = 0x40000000;    // tile_dim0=0x4000 -> bits[31:16]
  g1[4] = 0;             // tile_dim1=0 (unused), tile_dim2=0
  g1[5] = 0x4000;        // tensor_dim0_stride[31:0] = 16384
  g1[6] = 0;
  g1[7] = 0;
  const i32x4 z4 = {0, 0, 0, 0};
#if defined(__clang_major__) && __clang_major__ >= 23
  const i32x8 z8 = {0, 0, 0, 0, 0, 0, 0, 0};
  __builtin_amdgcn_tensor_load_to_lds(g0, g1, z4, z4, z8, 0);
#else
  __builtin_amdgcn_tensor_load_to_lds(g0, g1, z4, z4, 0);
#endif
  __builtin_amdgcn_s_wait_tensorcnt((short)0);
#else
  (void)gsrc; (void)lds_dst;
#endif
}

// Stage 16 contiguous rows (16x1024 f32) of src into ldsA; whole WG helper.
__device__ __forceinline__ void stage_panel(const float* src, float* ldsA,
                                            int t) {
#if USE_TDM
  if (t == 0) tdm_copy_16k_f32(src, ldsA);   // TDM ignores EXEC; wave0 issues
#else
  const float4* s4 = (const float4*)src;
  float4* d4 = (float4*)ldsA;
#pragma unroll
  for (int i = 0; i < 16; ++i) d4[t + i * 256] = s4[t + i * 256];
#endif
  __syncthreads();
}

// ---------------------------------------------------------------------------
// Projection GEMM: X[8192,1024] @ W[1024,1024] -> head-major [b,h,s,d]
// grid = (N/128, M/16) = (8, 512), block = 256 (8 waves, one 16x16 tile each)
// ---------------------------------------------------------------------------
__global__ __launch_bounds__(256) void proj_qkv_kernel(
    const float* __restrict__ X, const float* __restrict__ W,
    float* __restrict__ OutHead) {
  __shared__ float ldsA[16 * DM];  // 64 KB: 16 contiguous rows of X
  const int t = threadIdx.x;
  const int wave = t >> 5, lane = t & 31;
  const int rb = blockIdx.y;

  stage_panel(X + (size_t)rb * 16 * DM, ldsA, t);

  const int ct = blockIdx.x * 128 + wave * 16;  // output column tile base
  const int mn = lane & 15;                     // M for A-frag, N for B/C/D
  const int kh = lane >> 4;                     // K-half select
  v8f c = {};
#pragma unroll 4
  for (int k0 = 0; k0 < DM; k0 += 4) {
    const int ka = k0 + 2 * kh;
    v2f a = *(const v2f*)&ldsA[mn * DM + ka];
    v2f b;
    b.x = W[(size_t)ka * DM + ct + mn];        // B[k][n] (L2-resident weight)
    b.y = W[(size_t)(ka + 1) * DM + ct + mn];
    c = wmma4(a, b, c);
    if (((k0 & 127) == 0) && (k0 + 256 < DM))
      __builtin_prefetch(&W[(size_t)(k0 + 256) * DM + ct + mn], 0, 0);
  }
  // scatter into [b, h, s, d]
#pragma unroll
  for (int r = 0; r < 8; ++r) {
    const int m = r + 8 * kh;
    const int Row = rb * 16 + m;   // = b*1024 + s
    const int b_ = Row >> 10, s = Row & 1023;
    const int Col = ct + mn;       // = h*64 + d
    const int h = Col >> 6, d = Col & 63;
    OutHead[(((size_t)(b_ * NH + h)) * SEQ + s) * DK + d] = c[r];
  }
}

// ---------------------------------------------------------------------------
// Fused attention: scores -> softmax -> p_attn (global) -> P@V
// grid = (64 row-blocks, 16 heads, 8 batch), block = 128 (4 waves)
// ---------------------------------------------------------------------------
__global__ __launch_bounds__(128) void attn_kernel(
    const float* __restrict__ QH, const float* __restrict__ KH,
    const float* __restrict__ VH, float* __restrict__ P,
    float* __restrict__ O) {
  __shared__ float ldsS[16 * SEQ];  // exactly 64 KB score panel
  const int lane = threadIdx.x & 31, wave = threadIdx.x >> 5;
  const int mn = lane & 15, kh = lane >> 4;
  const int h = blockIdx.y, b = blockIdx.z;
  const int s0 = blockIdx.x * 16;
  const size_t headOff = ((size_t)b * NH + h) * SEQ * DK;

  // ---- phase 1: S = (Q @ K^T) * 1/sqrt(64) ----
  const float* Qb = QH + headOff + (size_t)s0 * DK;
  v2f aq[16];  // full 16x64 Q block held as A-fragments in registers
#pragma unroll
  for (int i = 0; i < 16; ++i)
    aq[i] = *(const v2f*)&Qb[mn * DK + i * 4 + 2 * kh];

  for (int tt = 0; tt < 16; ++tt) {  // 16 key tiles per wave (4 waves * 256 keys)
    const int keyBase = (wave * 16 + tt) * 16;
    const float* Kb = KH + headOff + (size_t)keyBase * DK;
    v8f c = {};
#pragma unroll
    for (int i = 0; i < 16; ++i) {
      // B[k][n] = K[keyBase+n][k] : transposed read, L2-resident (256 KB/head)
      v2f bf = *(const v2f*)&Kb[mn * DK + i * 4 + 2 * kh];
      c = wmma4(aq[i], bf, c);
    }
#pragma unroll
    for (int r = 0; r < 8; ++r)
      ldsS[(r + 8 * kh) * SEQ + keyBase + mn] = c[r] * 0.125f;
  }
  __syncthreads();

  // ---- phase 2: softmax (wave32 shuffle reductions, per-thread 128-col chunk)
  const int row = wave * 4 + (lane >> 3);  // 16 rows
  const int seg = lane & 7;                // 8 segments of 128 columns
  float4* chunk = (float4*)&ldsS[row * SEQ + seg * 128];
  float mx = -3.0e38f;
#pragma unroll
  for (int j = 0; j < 32; ++j) {
    float4 v = chunk[j];
    mx = fmaxf(mx, fmaxf(fmaxf(v.x, v.y), fmaxf(v.z, v.w)));
  }
  mx = fmaxf(mx, __shfl_xor(mx, 1, 32));
  mx = fmaxf(mx, __shfl_xor(mx, 2, 32));
  mx = fmaxf(mx, __shfl_xor(mx, 4, 32));
  float sum = 0.f;
#pragma unroll
  for (int j = 0; j < 32; ++j) {
    float4 v = chunk[j];
    v.x = __expf(v.x - mx); v.y = __expf(v.y - mx);
    v.z = __expf(v.z - mx); v.w = __expf(v.w - mx);
    chunk[j] = v;
    sum += (v.x + v.y) + (v.z + v.w);
  }
  sum += __shfl_xor(sum, 1, 32);
  sum += __shfl_xor(sum, 2, 32);
  sum += __shfl_xor(sum, 4, 32);
  const float inv = 1.0f / sum;
  // normalize in LDS and stream p_attn to HBM (512B contiguous per lane)
  float4* pg = (float4*)(P + (((size_t)(b * NH + h)) * SEQ + s0 + row) * SEQ +
                         seg * 128);
#pragma unroll
  for (int j = 0; j < 32; ++j) {
    float4 v = chunk[j];
    v.x *= inv; v.y *= inv; v.z *= inv; v.w *= inv;
    chunk[j] = v;
    pg[j] = v;
  }
  __syncthreads();

  // ---- phase 3: O = P @ V  (one 16x16 d-tile per wave, K = 1024 keys) ----
  const int dBase = wave * 16;
  const float* Vb = VH + headOff;
  v8f c = {};
#pragma unroll 4
  for (int k0 = 0; k0 < SEQ; k0 += 4) {
    const int ka = k0 + 2 * kh;
    v2f a = *(const v2f*)&ldsS[mn * SEQ + ka];
    v2f bf;
    bf.x = Vb[(size_t)ka * DK + dBase + mn];
    bf.y = Vb[(size_t)(ka + 1) * DK + dBase + mn];
    c = wmma4(a, bf, c);
  }
  // store concat-head layout [b*1024+s][h*64+d]
  float* Ob = O + (((size_t)b * SEQ) + s0) * DM + h * DK + dBase;
#pragma unroll
  for (int r = 0; r < 8; ++r)
    Ob[(size_t)(r + 8 * kh) * DM + mn] = c[r];
}

// ---------------------------------------------------------------------------
// FC GEMM + residual: A[8192,1024] @ Wfc[1024,1024] + R -> Out (row-major)
// ---------------------------------------------------------------------------
__global__ __launch_bounds__(256) void fc_kernel(
    const float* __restrict__ A, const float* __restrict__ W,
    const float* __restrict__ R, float* __restrict__ Out) {
  __shared__ float ldsA[16 * DM];
  const int t = threadIdx.x;
  const int wave = t >> 5, lane = t & 31;
  const int rb = blockIdx.y;

  stage_panel(A + (size_t)rb * 16 * DM, ldsA, t);

  const int ct = blockIdx.x * 128 + wave * 16;
  const int mn = lane & 15, kh = lane >> 4;
  v8f c = {};
#pragma unroll 4
  for (int k0 = 0; k0 < DM; k0 += 4) {
    const int ka = k0 + 2 * kh;
    v2f a = *(const v2f*)&ldsA[mn * DM + ka];
    v2f b;
    b.x = W[(size_t)ka * DM + ct + mn];
    b.y = W[(size_t)(ka + 1) * DM + ct + mn];
    c = wmma4(a, b, c);
    if (((k0 & 127) == 0) && (k0 + 256 < DM))
      __builtin_prefetch(&W[(size_t)(k0 + 256) * DM + ct + mn], 0, 0);
  }
#pragma unroll
  for (int r = 0; r < 8; ++r) {
    const int m = r + 8 * kh;
    const size_t idx = ((size_t)rb * 16 + m) * DM + ct + mn;
    Out[idx] = c[r] + R[idx];
  }
}

// ---------------------------------------------------------------------------
// Row-wise LayerNorm: one block per row of 1024
// ---------------------------------------------------------------------------
__global__ __launch_bounds__(256) void ln_kernel(
    const float* __restrict__ X, const float* __restrict__ gamma,
    const float* __restrict__ beta, float* __restrict__ Out) {
  __shared__ float red[2][8];
  const int t = threadIdx.x, lane = t & 31, wave = t >> 5;
  const float* xr = X + (size_t)blockIdx.x * DM;
  float s1 = 0.f, s2 = 0.f;
  for (int j = t; j < DM; j += 256) {
    float x = xr[j];
    s1 += x;
    s2 += x * x;
  }
  for (int off = 16; off; off >>= 1) {
    s1 += __shfl_xor(s1, off, 32);
    s2 += __shfl_xor(s2, off, 32);
  }
  if (lane == 0) { red[0][wave] = s1; red[1][wave] = s2; }
  __syncthreads();
  float S1 = 0.f, S2 = 0.f;
#pragma unroll
  for (int i = 0; i < 8; ++i) { S1 += red[0][i]; S2 += red[1][i]; }
  const float mean = S1 * (1.0f / DM);
  const float var = S2 * (1.0f / DM) - mean * mean;
  const float rstd = rsqrtf(var + 1e-6f);
  float* orow = Out + (size_t)blockIdx.x * DM;
  for (int j = t; j < DM; j += 256)
    orow[j] = (xr[j] - mean) * rstd * gamma[j] + beta[j];
}

// ---------------------------------------------------------------------------
extern "C" void kernel_launch(void* const* d_in, const int* in_sizes, int n_in,
                              void* d_out, int out_size, void* d_ws,
                              size_t ws_size, hipStream_t stream) {
  (void)in_sizes; (void)n_in; (void)out_size; (void)ws_size;
  const float* q     = (const float*)d_in[0];
  const float* k     = (const float*)d_in[1];
  const float* v     = (const float*)d_in[2];
  const float* w_q   = (const float*)d_in[3];
  const float* w_k   = (const float*)d_in[4];
  const float* w_v   = (const float*)d_in[5];
  const float* w_fc  = (const float*)d_in[6];
  const float* gamma = (const float*)d_in[7];
  const float* beta  = (const float*)d_in[8];

  float* out   = (float*)d_out;                       // [8,1024,1024]
  float* pattn = out + (size_t)BS * SEQ * DM;         // [8,16,1024,1024]

  float* ws = (float*)d_ws;
  const size_t SZ = (size_t)BS * NH * SEQ * DK;  // 8.39M floats (32 MB)
  float* qh = ws;           // [b,h,s,d]
  float* kh = ws + SZ;
  float* vh = ws + 2 * SZ;
  float* ao = ws + 3 * SZ;  // attention output, concat-head [8192,1024]
  float* fo = ws;           // fc output reuses qh slot (stream-ordered)

  dim3 gG(8, 512), bG(256);
  proj_qkv_kernel<<<gG, bG, 0, stream>>>(q, w_q, qh);
  proj_qkv_kernel<<<gG, bG, 0, stream>>>(k, w_k, kh);
  proj_qkv_kernel<<<gG, bG, 0, stream>>>(v, w_v, vh);
  attn_kernel<<<dim3(64, 16, 8), 128, 0, stream>>>(qh, kh, vh, pattn, ao);
  fc_kernel<<<gG, bG, 0, stream>>>(ao, w_fc, q, fo);
  ln_kernel<<<BS * SEQ, 256, 0, stream>>>(fo, gamma, beta, out);
}